// ModelNew_80908593922815
// MI455X (gfx1250) — compile-verified
//
#include <hip/hip_runtime.h>

typedef __attribute__((ext_vector_type(2))) float v2f;
typedef __attribute__((ext_vector_type(8))) float v8f;

#define CI      8
#define OC      16
#define KTOT    216          // CI * 27
#define NSTEP   54           // KTOT / 4  (K per WMMA = 4)
#define D_IN    32
#define H_IN    64
#define W_IN    64
#define DP      15           // pooled depth
#define HP      31           // pooled height
#define WP      31           // pooled width
#define NPOS    248          // 2*2*62 conv outputs per slab
#define OUT_STRIDE 17        // 16 + 1 pad to dodge bank conflicts
#define POOL_COUNT 14415.0f  // 15*31*31

#if defined(__has_builtin)
#  if __has_builtin(__builtin_amdgcn_global_load_async_to_lds_b32)
#    define HAVE_ASYNC_LDS 1
#  endif
#endif
#ifndef HAVE_ASYNC_LDS
#  define HAVE_ASYNC_LDS 0
#endif

__device__ __forceinline__ void wait_async_then_barrier()
{
#if HAVE_ASYNC_LDS
#  if __has_builtin(__builtin_amdgcn_s_wait_asynccnt)
    __builtin_amdgcn_s_wait_asynccnt(0);
#  else
    asm volatile("s_wait_asynccnt 0x0" ::: "memory");
#  endif
#endif
    __syncthreads();
}

// Input window in LDS, ci-innermost so WMMA A-pairs are contiguous b64 loads:
//   xs2[d''(4)][h''(4)][w''(64)][ci(8)]  -> addr = d*2048 + h*512 + w*8 + ci
// Weights pre-permuted into per-step pair layout:
//   wl2[(kk*32 + half*16 + n)*2 + s] = W[k'=4kk+2half+s][n]
// with K-order k' = ((kd*3+kh)*3+kw)*8 + ci.

__global__ __launch_bounds__(256) void conv_pool_wmma_kernel(
    const float* __restrict__ x,      // (32,8,32,64,64)
    const float* __restrict__ wgt,    // (16,8,3,3,3)
    const float* __restrict__ cbias,  // (16,)
    float* __restrict__ accum)        // (32,16) partial pooled sums
{
    __shared__ float xs2[4 * 4 * 64 * CI];   // 8192 f
    __shared__ float wl2[KTOT * OC];         // 3456 f
    __shared__ float outL[NPOS * OUT_STRIDE];
    __shared__ float cb[OC];
    __shared__ float chsum[OC];

    const int tid = threadIdx.x;
    int blk = blockIdx.x;
    const int hp = blk % HP; blk /= HP;
    const int dp = blk % DP; blk /= DP;
    const int b  = blk;                    // 0..31
    const int d0 = 2 * dp, h0 = 2 * hp;

#if HAVE_ASYNC_LDS
    // ---- Async global->LDS staging (no VGPR round-trip, ASYNCcnt-tracked)
    #pragma unroll 4
    for (int e = tid; e < 8192; e += 256) {
        const int r  = e >> 6;             // row (ci,dz,hz) 0..127
        const int w  = e & 63;
        const int ci = r >> 4;
        const int dz = (r >> 2) & 3;
        const int hz = r & 3;
        const size_t gaddr = ((((size_t)b * CI + ci) * D_IN + (d0 + dz)) * H_IN
                              + (h0 + hz)) * W_IN + w;
        const int laddr = dz * 2048 + hz * 512 + w * 8 + ci;
        __builtin_amdgcn_global_load_async_to_lds_b32(
            (int*)(x + gaddr), (int*)&xs2[laddr], 0, 0);
    }
    for (int i = tid; i < KTOT * OC; i += 256) {
        const int kp  = i >> 4;
        const int n   = i & 15;
        const int tap = kp >> 3;
        const int ci  = kp & 7;
        const int kd  = tap / 9;
        const int kh  = (tap / 3) % 3;
        const int kw  = tap % 3;
        const int Ko  = ci * 27 + kd * 9 + kh * 3 + kw;
        const int kk  = kp >> 2;
        const int hf  = (kp >> 1) & 1;
        const int s   = kp & 1;
        __builtin_amdgcn_global_load_async_to_lds_b32(
            (int*)(wgt + n * KTOT + Ko),
            (int*)&wl2[(kk * 32 + hf * 16 + n) * 2 + s], 0, 0);
    }
#else
    // ---- Fallback staging through VGPRs
    for (int i = tid; i < 2048; i += 256) {
        const int r   = i >> 4;
        const int seg = i & 15;
        const int ci  = r >> 4;
        const int dz  = (r >> 2) & 3;
        const int hz  = r & 3;
        const size_t gaddr = ((((size_t)b * CI + ci) * D_IN + (d0 + dz)) * H_IN
                              + (h0 + hz)) * W_IN + seg * 4;
        const float4 v = *(const float4*)(x + gaddr);
        const int base = dz * 2048 + hz * 512 + seg * 32 + ci;
        xs2[base]      = v.x;
        xs2[base + 8]  = v.y;
        xs2[base + 16] = v.z;
        xs2[base + 24] = v.w;
    }
    for (int i = tid; i < KTOT * OC; i += 256) {
        const int kp  = i >> 4;
        const int n   = i & 15;
        const int tap = kp >> 3;
        const int ci  = kp & 7;
        const int kd  = tap / 9;
        const int kh  = (tap / 3) % 3;
        const int kw  = tap % 3;
        const int Ko  = ci * 27 + kd * 9 + kh * 3 + kw;
        const int kk  = kp >> 2;
        const int hf  = (kp >> 1) & 1;
        const int s   = kp & 1;
        wl2[(kk * 32 + hf * 16 + n) * 2 + s] = wgt[n * KTOT + Ko];
    }
#endif
    if (tid < OC) { cb[tid] = cbias[tid]; chsum[tid] = 0.0f; }
    wait_async_then_barrier();

    // ---- Implicit-GEMM conv via V_WMMA_F32_16X16X4_F32 (all offsets immediate)
    const int lane = tid & 31;
    const int wid  = tid >> 5;              // wave 0..7
    const int n    = lane & 15;             // output channel
    const int half = lane >> 4;             // 0/1: which K sub-pair this lane holds
    const int m    = lane & 15;             // A row / position within tile

    int abase2[2];
    #pragma unroll
    for (int tt = 0; tt < 2; ++tt) {
        int q = (wid + tt * 8) * 16 + m;
        if (q > NPOS - 1) q = NPOS - 1;     // clamp addr for dead positions
        const int dd = q / 124;
        const int r2 = q - dd * 124;
        const int hh = r2 / 62;
        const int w  = r2 - hh * 62;
        abase2[tt] = dd * 2048 + hh * 512 + w * 8 + half * 2;
    }
    const int bbase = (half * 16 + n) * 2;

    v8f acc0 = {0.f, 0.f, 0.f, 0.f, 0.f, 0.f, 0.f, 0.f};
    v8f acc1 = {0.f, 0.f, 0.f, 0.f, 0.f, 0.f, 0.f, 0.f};

    #pragma unroll
    for (int kk = 0; kk < NSTEP; ++kk) {
        const int tap  = kk >> 1;
        const int kd   = tap / 9;
        const int kh   = (tap / 3) % 3;
        const int kw   = tap % 3;
        const int aofs = kd * 2048 + kh * 512 + kw * 8 + (kk & 1) * 4; // constant
        const v2f bm = *(const v2f*)(&wl2[bbase + kk * 64]);           // b64
        const v2f a0 = *(const v2f*)(&xs2[abase2[0] + aofs]);          // b64
        const v2f a1 = *(const v2f*)(&xs2[abase2[1] + aofs]);          // b64
        acc0 = __builtin_amdgcn_wmma_f32_16x16x4_f32(
            false, a0, false, bm, (short)0, acc0, false, false);
        acc1 = __builtin_amdgcn_wmma_f32_16x16x4_f32(
            false, a1, false, bm, (short)0, acc1, false, false);
    }

    // ---- Epilogue: + conv_bias, / DIVISOR, scatter to LDS tile
    const int mhi = half * 8;
    #pragma unroll
    for (int v = 0; v < 8; ++v) {
        const int q0 = wid * 16 + (v + mhi);
        const int q1 = (wid + 8) * 16 + (v + mhi);
        outL[q0 * OUT_STRIDE + n] = (acc0[v] + cb[n]) * 0.5f;
        if (q1 < NPOS)
            outL[q1 * OUT_STRIDE + n] = (acc1[v] + cb[n]) * 0.5f;
    }
    __syncthreads();

    // ---- MaxPool 2x2x2 + per-channel sum (31 pool cols x 16 ch = 496 items)
    for (int i = tid; i < WP * OC; i += 256) {
        const int wq = i >> 4;
        const int c  = i & 15;
        float mx = -3.402823466e38f;
        #pragma unroll
        for (int dd = 0; dd < 2; ++dd)
            #pragma unroll
            for (int hh = 0; hh < 2; ++hh)
                #pragma unroll
                for (int ww = 0; ww < 2; ++ww) {
                    const int q = dd * 124 + hh * 62 + 2 * wq + ww;
                    mx = fmaxf(mx, outL[q * OUT_STRIDE + c]);
                }
        atomicAdd(&chsum[c], mx);          // ds_add_f32
    }
    __syncthreads();
    if (tid < OC) atomicAdd(&accum[b * OC + tid], chsum[tid]);  // global_atomic_add_f32
}

// out[b] = sum_c( pooled_sum[b][c]/14415 + bias[c] )
__global__ void finalize_kernel(const float* __restrict__ accum,
                                const float* __restrict__ bias,
                                float* __restrict__ out)
{
    const int b = threadIdx.x;
    if (b < 32) {
        float s = 0.0f;
        for (int c = 0; c < OC; ++c)
            s += accum[b * OC + c] * (1.0f / POOL_COUNT) + bias[c];
        out[b] = s;
    }
}

extern "C" void kernel_launch(void* const* d_in, const int* in_sizes, int n_in,
                              void* d_out, int out_size, void* d_ws, size_t ws_size,
                              hipStream_t stream)
{
    const float* x     = (const float*)d_in[0];
    const float* wgt   = (const float*)d_in[1];
    const float* cbias = (const float*)d_in[2];
    const float* bias  = (const float*)d_in[3];
    float* out   = (float*)d_out;
    float* accum = (float*)d_ws;           // 32*16 floats

    (void)hipMemsetAsync(accum, 0, 32 * OC * sizeof(float), stream);

    const int nblocks = 32 * DP * HP;      // 14880 slabs
    conv_pool_wmma_kernel<<<nblocks, 256, 0, stream>>>(x, wgt, cbias, accum);
    finalize_kernel<<<1, 32, 0, stream>>>(accum, bias, out);
}